// VectorQuantizer_56590489092605
// MI455X (gfx1250) — compile-verified
//
#include <hip/hip_runtime.h>

#define DIM   256
#define NEMB  1024
#define DECAYF 0.99f
#define OMDF   0.01f
#define EPSV   1e-5f

typedef __attribute__((ext_vector_type(2))) float v2f;
typedef __attribute__((ext_vector_type(8))) float v8f;

// ---------------------------------------------------------------- prep ----
__global__ void vq_zero(float* __restrict__ p, int n) {
    int i = blockIdx.x * blockDim.x + threadIdx.x;
    if (i < n) p[i] = 0.0f;
}

// embedT[j*DIM + d] = embed[d*NEMB + j]
__global__ void vq_transpose(const float* __restrict__ embed,
                             float* __restrict__ embedT) {
    int i = blockIdx.x * blockDim.x + threadIdx.x;   // 0 .. DIM*NEMB-1
    int j = i >> 8;          // code index
    int d = i & (DIM - 1);   // dim index
    embedT[(size_t)j * DIM + d] = embed[(size_t)d * NEMB + j];
}

// e2[j] = sum_d embed[d][j]^2  (read from transposed copy, contiguous)
__global__ void vq_colnorm(const float* __restrict__ embedT,
                           float* __restrict__ e2) {
    int j = blockIdx.x * blockDim.x + threadIdx.x;   // 0 .. NEMB-1
    const float* col = embedT + (size_t)j * DIM;
    float s = 0.0f;
    for (int d = 0; d < DIM; ++d) { float v = col[d]; s += v * v; }
    e2[j] = s;
}

// Pre-swizzle codebook into WMMA f32 16x16x4 B-fragment order.
// Element (float4) index: ((T*16 + g)*2 + p)*32 + lane
//   T = N-tile (0..63), g = K-group of 16 (0..15), p = pair (0..1), lane 0..31
//   h = lane>>4, n = T*16 + (lane&15), k0 = g*16 + p*8
//   value = { E[k0+2h][n], E[k0+2h+1][n], E[k0+4+2h][n], E[k0+4+2h+1][n] }
__global__ void vq_bfrag(const float* __restrict__ embed,
                         float4* __restrict__ bf) {
    int i = blockIdx.x * blockDim.x + threadIdx.x;   // 0 .. 65535
    int lane = i & 31;
    int t2   = i >> 5;
    int p    = t2 & 1;
    int g    = (t2 >> 1) & 15;
    int T    = t2 >> 5;
    int h    = lane >> 4;
    int n    = T * 16 + (lane & 15);
    int k0   = g * 16 + p * 8;
    float4 v;
    v.x = embed[(size_t)(k0 + 2 * h)     * NEMB + n];
    v.y = embed[(size_t)(k0 + 2 * h + 1) * NEMB + n];
    v.z = embed[(size_t)(k0 + 4 + 2 * h)     * NEMB + n];
    v.w = embed[(size_t)(k0 + 4 + 2 * h + 1) * NEMB + n];
    bf[i] = v;
}

static __device__ __forceinline__ v8f wmma_f32_k4(float ax, float ay,
                                                  float bx, float by, v8f c) {
    v2f a, b;
    a.x = ax; a.y = ay;
    b.x = bx; b.y = by;
    return __builtin_amdgcn_wmma_f32_16x16x4_f32(
        false, a, false, b, (short)0, c, false, false);
}

// ------------------------------------------------- distance + argmin ------
// grid = n_tok/16 blocks, 256 threads (8 waves of 32).
// Block owns 16 token rows; wave w owns codes [w*128, w*128+128).
__global__ __launch_bounds__(256) void vq_dist_argmin(
    const float* __restrict__ x,        // [n_tok, DIM]
    const float4* __restrict__ bf,      // swizzled codebook fragments
    const float* __restrict__ embedT,   // [NEMB, DIM]
    const float* __restrict__ e2,       // [NEMB]
    float* __restrict__ out_q,          // [n_tok, DIM]
    float* __restrict__ out_ind,        // [n_tok] (as float)
    float* __restrict__ counts,         // [NEMB]
    float* __restrict__ esum)           // [DIM, NEMB]
{
    __shared__ float  tileA[16][DIM];   // 16 KB raw token tile
    __shared__ float4 af[16 * 2 * 32];  // 16 KB A fragments, index (g*2+p)*32+lane
    __shared__ float  x2[16];
    __shared__ float  wminv[8][16];
    __shared__ int    wmini[8][16];
    __shared__ int    fidx[16];

    const int t    = threadIdx.x;
    const int row0 = blockIdx.x * 16;

    // cooperative load of the 16x256 A tile (coalesced)
    #pragma unroll
    for (int i = 0; i < 16; ++i) {
        int l = i * 256 + t;
        int r = l >> 8, d = l & (DIM - 1);
        tileA[r][d] = x[(size_t)(row0 + r) * DIM + d];
    }
    __syncthreads();

    // row norms
    if (t < 16) {
        float s = 0.0f;
        for (int d = 0; d < DIM; ++d) { float v = tileA[t][d]; s += v * v; }
        x2[t] = s;
    }

    // swizzle A into fragment order: 1024 float4, 4 per thread
    #pragma unroll
    for (int i = 0; i < 4; ++i) {
        int fi   = i * 256 + t;
        int ln   = fi & 31;
        int p    = (fi >> 5) & 1;
        int g    = fi >> 6;
        int h    = ln >> 4;
        int m    = ln & 15;
        int k0   = g * 16 + p * 8;
        float4 v;
        v.x = tileA[m][k0 + 2 * h];
        v.y = tileA[m][k0 + 2 * h + 1];
        v.z = tileA[m][k0 + 4 + 2 * h];
        v.w = tileA[m][k0 + 4 + 2 * h + 1];
        af[fi] = v;
    }
    __syncthreads();

    const int wave = t >> 5;
    const int lane = t & 31;
    const int half = lane >> 4;   // 0: lanes 0-15, 1: lanes 16-31
    const int l16  = lane & 15;

    // hoist x2 (rows this lane owns in the C tile) and e2 (cols per N-tile)
    float xs[8];
    #pragma unroll
    for (int i = 0; i < 8; ++i) xs[i] = x2[i + 8 * half];
    float e2v[8];
    #pragma unroll
    for (int nt = 0; nt < 8; ++nt)
        e2v[nt] = e2[wave * 128 + nt * 16 + l16];

    float rminv[8];
    int   rmini[8];
    #pragma unroll
    for (int i = 0; i < 8; ++i) { rminv[i] = 3.4e38f; rmini[i] = 0; }

    // 8 N-tiles of 16 codes each per wave
    for (int nt = 0; nt < 8; ++nt) {
        const int T = wave * 8 + nt;
        const int n = T * 16 + l16;
        const float4* bbase = bf + ((size_t)T * 16) * 64 + lane;

        v8f c = {};
        // K loop: 16 groups x 4 WMMA; 2 ds_b128 + 2 global_b128 per group
        #pragma unroll 4
        for (int g = 0; g < 16; ++g) {
            float4 a0 = af[(g * 2 + 0) * 32 + lane];
            float4 a1 = af[(g * 2 + 1) * 32 + lane];
            float4 b0 = bbase[(size_t)g * 64];        // p = 0
            float4 b1 = bbase[(size_t)g * 64 + 32];   // p = 1
            c = wmma_f32_k4(a0.x, a0.y, b0.x, b0.y, c);   // k = g*16
            c = wmma_f32_k4(a0.z, a0.w, b0.z, b0.w, c);   // k = g*16+4
            c = wmma_f32_k4(a1.x, a1.y, b1.x, b1.y, c);   // k = g*16+8
            c = wmma_f32_k4(a1.z, a1.w, b1.z, b1.w, c);   // k = g*16+12
        }

        // dist[m][n] = x2[m] - 2*dot + e2[n]; lane holds col n, rows i+8*half
        #pragma unroll
        for (int i = 0; i < 8; ++i) {
            float d = xs[i] - 2.0f * c[i] + e2v[nt];
            if (d < rminv[i]) { rminv[i] = d; rmini[i] = n; }
        }
    }

    // reduce (min,argmin) across the 16 lanes sharing each row (wave32 shuffles)
    #pragma unroll
    for (int i = 0; i < 8; ++i) {
        float v  = rminv[i];
        int   id = rmini[i];
        #pragma unroll
        for (int off = 1; off < 16; off <<= 1) {
            float ov = __shfl_xor(v, off, 32);
            int   oi = __shfl_xor(id, off, 32);
            if (ov < v || (ov == v && oi < id)) { v = ov; id = oi; }
        }
        if (l16 == 0) { wminv[wave][i + 8 * half] = v; wmini[wave][i + 8 * half] = id; }
    }
    __syncthreads();

    // reduce across the 8 waves; first-index tie-break matches jnp.argmin
    if (t < 16) {
        float v  = wminv[0][t];
        int   id = wmini[0][t];
        #pragma unroll
        for (int w = 1; w < 8; ++w) {
            float ov = wminv[w][t];
            int   oi = wmini[w][t];
            if (ov < v || (ov == v && oi < id)) { v = ov; id = oi; }
        }
        fidx[t] = id;
        out_ind[row0 + t] = (float)id;
        atomicAdd(&counts[id], 1.0f);
    }
    __syncthreads();

    // quantize gather (coalesced via embedT) + embed_sum scatter (fp32 L2 atomics)
    for (int r = 0; r < 16; ++r) {
        const int   id = fidx[r];
        const float q  = embedT[(size_t)id * DIM + t];
        const float xv = tileA[r][t];
        out_q[(size_t)(row0 + r) * DIM + t] = xv + (q - xv);  // straight-through value
        atomicAdd(&esum[(size_t)t * NEMB + id], xv);
    }
}

// --------------------------------------------------------- epilogues ------
__global__ __launch_bounds__(1024) void vq_cluster(
    const float* __restrict__ cluster_size,
    const float* __restrict__ counts,
    float* __restrict__ out_ncs,
    float* __restrict__ smoothed)
{
    __shared__ float red[NEMB];
    int j = threadIdx.x;
    float ncs = cluster_size[j] * DECAYF + OMDF * counts[j];
    out_ncs[j] = ncs;
    red[j] = ncs;
    __syncthreads();
    for (int s = NEMB / 2; s > 0; s >>= 1) {
        if (j < s) red[j] += red[j + s];
        __syncthreads();
    }
    float S = red[0];
    smoothed[j] = (ncs + EPSV) / (S + (float)NEMB * EPSV) * S;
}

__global__ void vq_embed_update(
    const float* __restrict__ embed_avg,
    const float* __restrict__ esum,
    const float* __restrict__ smoothed,
    float* __restrict__ out_nea,
    float* __restrict__ out_norm)
{
    int i = blockIdx.x * blockDim.x + threadIdx.x;   // 0 .. DIM*NEMB-1
    int j = i & (NEMB - 1);
    float nea = embed_avg[i] * DECAYF + OMDF * esum[i];
    out_nea[i]  = nea;
    out_norm[i] = nea / smoothed[j];
}

// ------------------------------------------------------------ launch ------
extern "C" void kernel_launch(void* const* d_in, const int* in_sizes, int n_in,
                              void* d_out, int out_size, void* d_ws, size_t ws_size,
                              hipStream_t stream) {
    const float* x     = (const float*)d_in[0];   // [8,4096,256]
    const float* embed = (const float*)d_in[1];   // [256,1024]
    const float* cs    = (const float*)d_in[2];   // [1024]
    const float* ea    = (const float*)d_in[3];   // [256,1024]
    const int n_tok = in_sizes[0] / DIM;          // 32768

    // output layout: concat in return order
    float* out      = (float*)d_out;
    float* out_q    = out;                               // n_tok*DIM
    float* out_ind  = out_q   + (size_t)n_tok * DIM;     // n_tok
    float* out_norm = out_ind + n_tok;                   // DIM*NEMB
    float* out_ncs  = out_norm + (size_t)DIM * NEMB;     // NEMB
    float* out_nea  = out_ncs  + NEMB;                   // DIM*NEMB

    // workspace layout (~3.1 MB)
    float* ws       = (float*)d_ws;
    float* counts   = ws;                                // NEMB
    float* esum     = counts   + NEMB;                   // DIM*NEMB
    float* e2       = esum     + (size_t)DIM * NEMB;     // NEMB
    float* smoothed = e2       + NEMB;                   // NEMB
    float* embedT   = smoothed + NEMB;                   // NEMB*DIM
    float4* bfrag   = (float4*)(embedT + (size_t)NEMB * DIM); // 65536 float4

    const int nzero = NEMB + DIM * NEMB;
    vq_zero      <<<(nzero + 255) / 256, 256, 0, stream>>>(counts, nzero);
    vq_transpose <<<(DIM * NEMB) / 256,  256, 0, stream>>>(embed, embedT);
    vq_colnorm   <<<NEMB / 256,          256, 0, stream>>>(embedT, e2);
    vq_bfrag     <<<(DIM * NEMB / 4) / 256, 256, 0, stream>>>(embed, bfrag);
    vq_dist_argmin<<<n_tok / 16,         256, 0, stream>>>(x, bfrag, embedT, e2,
                                                           out_q, out_ind, counts, esum);
    vq_cluster   <<<1, NEMB, 0, stream>>>(cs, counts, out_ncs, smoothed);
    vq_embed_update<<<(DIM * NEMB) / 256, 256, 0, stream>>>(ea, esum, smoothed,
                                                            out_nea, out_norm);
}